// GraphAttentionLayer_90340342104879
// MI455X (gfx1250) — compile-verified
//
#include <hip/hip_runtime.h>

// Problem constants (from reference)
constexpr int kB  = 8;
constexpr int kN  = 1024;
constexpr int kF  = 256;   // F_IN
constexpr int kH  = 8;
constexpr int kD  = 64;
constexpr int kHD = kH * kD;     // 512
constexpr int kBH = kB * kH;     // 64
constexpr int kNW = kN / 16;     // adjacency halfwords per row (64)

typedef __attribute__((ext_vector_type(2))) float v2f;
typedef __attribute__((ext_vector_type(8))) float v8f;

__device__ __forceinline__ v8f wmma_f32(v2f a, v2f b, v8f c) {
  // D(16x16,f32) += A(16x4,f32) * B(4x16,f32) + C
  return __builtin_amdgcn_wmma_f32_16x16x4_f32(false, a, false, b, (short)0, c,
                                               false, false);
}

// CDNA5 async copy: per-lane 16B global -> LDS, tracked by ASYNCcnt.
__device__ __forceinline__ void async_load_b128(unsigned lds_byte_off,
                                                unsigned long long gaddr) {
  asm volatile("global_load_async_to_lds_b128 %0, %1, off"
               :: "v"(lds_byte_off), "v"(gaddr) : "memory");
}
__device__ __forceinline__ void wait_async_le1() {
  asm volatile("s_wait_asynccnt 0x1" ::: "memory");
}
__device__ __forceinline__ void wait_async_0() {
  asm volatile("s_wait_asynccnt 0x0" ::: "memory");
}

// ---------------- K0: per-row inverse L2 norm (tf.nn.l2_normalize eps) -----
__global__ void k_norm(const float* __restrict__ h, float* __restrict__ inv) {
  int wave = threadIdx.x >> 5, lane = threadIdx.x & 31;
  int row = blockIdx.x * 8 + wave;              // row in [0, B*N)
  const float* p = h + (size_t)row * kF;
  float s = 0.f;
#pragma unroll
  for (int t = 0; t < kF / 32; ++t) { float x = p[lane + t * 32]; s += x * x; }
#pragma unroll
  for (int o = 16; o > 0; o >>= 1) s += __shfl_xor(s, o, 32);
  if (lane == 0) inv[row] = rsqrtf(fmaxf(s, 1e-12f));
}

// ---------------- K1: ht = h @ W  (8192x256 @ 256x512), fp32 WMMA ----------
__global__ void __launch_bounds__(256) k_proj(const float* __restrict__ h,
                                              const float* __restrict__ W,
                                              float* __restrict__ ht) {
  int wave = threadIdx.x >> 5, lane = threadIdx.x & 31;
  int wi = wave & 3, wj = wave >> 2;            // 4x2 wave grid -> WG tile 64x128
  int row0 = blockIdx.x * 64 + wi * 16;
  int col0 = blockIdx.y * 128 + wj * 64;
  int m = lane & 15, kb = (lane >> 4) * 2, nn = lane & 15;
  const float* Ar = h + (size_t)(row0 + m) * kF;
  v8f acc[4] = {};
  for (int k0 = 0; k0 < kF; k0 += 4) {
    v2f af = *(const v2f*)(Ar + k0 + kb);       // A[m][kb..kb+1], 8B aligned
    const float* Wr = W + (size_t)(k0 + kb) * kHD + col0 + nn;
#pragma unroll
    for (int t = 0; t < 4; ++t) {
      v2f bf; bf.x = Wr[t * 16]; bf.y = Wr[kHD + t * 16];
      acc[t] = wmma_f32(af, bf, acc[t]);
    }
  }
  int rb = lane >> 4;
#pragma unroll
  for (int t = 0; t < 4; ++t)
#pragma unroll
    for (int v = 0; v < 8; ++v)
      ht[(size_t)(row0 + rb * 8 + v) * kHD + col0 + t * 16 + nn] = acc[t][v];
}

// ---------------- K1b: src/tgt = ht . a  (one wave per (b,h,n)) ------------
__global__ void k_srctgt(const float* __restrict__ ht, const float* __restrict__ a,
                         float* __restrict__ src, float* __restrict__ tgt) {
  int wave = threadIdx.x >> 5, lane = threadIdx.x & 31;
  int r = blockIdx.x * 8 + wave;                // (b*H + h)*N + n
  int hh = (r >> 10) & 7;
  int b  = r >> 13;
  int n  = r & 1023;
  const float* base = ht + (size_t)(b * kN + n) * kHD + hh * kD;
  const float* av   = a + hh * 2 * kD;
  float s = base[lane] * av[lane]        + base[lane + 32] * av[lane + 32];
  float t = base[lane] * av[kD + lane]   + base[lane + 32] * av[kD + lane + 32];
#pragma unroll
  for (int o = 16; o > 0; o >>= 1) {
    s += __shfl_xor(s, o, 32);
    t += __shfl_xor(t, o, 32);
  }
  if (lane == 0) { src[r] = s; tgt[r] = t; }
}

// ---------------- K2: adjacency bitmask via fp32 WMMA + ballot -------------
__global__ void __launch_bounds__(256) k_adj(const float* __restrict__ h,
                                             const float* __restrict__ inv,
                                             unsigned short* __restrict__ adjh) {
  int wave = threadIdx.x >> 5, lane = threadIdx.x & 31;
  int w = blockIdx.x * 8 + wave;                // 16x16 tile id
  int b = w >> 12;                              // 4096 tiles per batch
  int rem = w & 4095;
  int i0 = (rem >> 6) * 16;
  int j0 = (rem & 63) * 16;
  const float* hb = h + (size_t)b * kN * kF;
  int m = lane & 15, kb = (lane >> 4) * 2, nn = lane & 15;
  const float* Ar = hb + (size_t)(i0 + m) * kF;
  const float* Br = hb + (size_t)(j0 + nn) * kF;   // B[k][n] = h[j0+n][k]
  v8f c = {};
  for (int k0 = 0; k0 < kF; k0 += 4) {
    v2f af = *(const v2f*)(Ar + k0 + kb);
    v2f bf = *(const v2f*)(Br + k0 + kb);
    c = wmma_f32(af, bf, c);
  }
  float invn = inv[b * kN + j0 + nn];
  int rb = lane >> 4;
#pragma unroll
  for (int v = 0; v < 8; ++v) {
    float sim = c[v] * inv[b * kN + i0 + rb * 8 + v] * invn;
    unsigned bal = __builtin_amdgcn_ballot_w32(sim > 0.5f);
    if (lane == 0)
      adjh[(size_t)(b * kN + i0 + v) * kNW + (j0 >> 4)] = (unsigned short)(bal & 0xffffu);
    else if (lane == 16)
      adjh[(size_t)(b * kN + i0 + 8 + v) * kNW + (j0 >> 4)] = (unsigned short)(bal >> 16);
  }
}

// ---------------- K3: row max & exp-sum (pass 1 of softmax) ----------------
__global__ void k_stats(const float* __restrict__ src, const float* __restrict__ tgt,
                        const unsigned short* __restrict__ adjh,
                        float* __restrict__ mrow, float* __restrict__ lrow) {
  int wave = threadIdx.x >> 5, lane = threadIdx.x & 31;
  int r = blockIdx.x * 8 + wave;                // (b*H+h)*N + i
  int b = r >> 13;
  int i = r & 1023;
  float s = src[r];
  const float* tg = tgt + (r & ~1023);
  const unsigned short* arow = adjh + (size_t)(b * kN + i) * kNW;
  float mx = -1.0e9f;
  for (int j = lane; j < kN; j += 32) {
    float e = s + tg[j];
    e = e > 0.f ? e : 0.2f * e;
    float em = ((arow[j >> 4] >> (j & 15)) & 1) ? e : -1.0e9f;
    mx = fmaxf(mx, em);
  }
#pragma unroll
  for (int o = 16; o > 0; o >>= 1) mx = fmaxf(mx, __shfl_xor(mx, o, 32));
  float l = 0.f;
  for (int j = lane; j < kN; j += 32) {
    float e = s + tg[j];
    e = e > 0.f ? e : 0.2f * e;
    float em = ((arow[j >> 4] >> (j & 15)) & 1) ? e : -1.0e9f;
    l += __expf(em - mx);
  }
#pragma unroll
  for (int o = 16; o > 0; o >>= 1) l += __shfl_xor(l, o, 32);
  if (lane == 0) { mrow[r] = mx; lrow[r] = l; }
}

// ---------------- K4: out = softmax(P) @ V, fused, fp32 WMMA ---------------
// One wave = 16 rows x 64 cols (4 accum tiles). WG = 8 waves = 128 rows of
// one (b,head). V j-tiles (16x64) double-buffered in LDS via
// GLOBAL_LOAD_ASYNC_TO_LDS_B128 (ASYNCcnt) so the next tile's memory latency
// hides under the current tile's 16 WMMAs. Stride 68 kills bank conflicts.
// P fragments are generated per-lane (disjoint element sets) -> no shuffle.
__global__ void __launch_bounds__(256) k_attn(
    const float* __restrict__ ht, const float* __restrict__ src,
    const float* __restrict__ tgt, const unsigned short* __restrict__ adjh,
    const float* __restrict__ mrow, const float* __restrict__ lrow,
    float* __restrict__ out) {
  __shared__ float Vt[2][16 * 68];
  int wave = threadIdx.x >> 5, lane = threadIdx.x & 31;
  int bh = blockIdx.x >> 3;                     // b*8 + h
  int b = bh >> 3, hh = bh & 7;
  int i0 = (blockIdx.x & 7) * 128 + wave * 16;
  int m = lane & 15, kb = (lane >> 4) * 2, nn = lane & 15;
  float s_i = src[bh * kN + i0 + m];
  float m_i = mrow[bh * kN + i0 + m];
  const unsigned short* arow = adjh + (size_t)(b * kN + i0 + m) * kNW;
  const float* tg = tgt + bh * kN;
  v8f acc[4] = {};

  // This thread's staging slot: row lr (0..15), 4 floats at col lc.
  int lr = threadIdx.x >> 4;
  int lc = (threadIdx.x & 15) * 4;
  // LDS byte offsets (generic LDS pointer low 32 bits == LDS offset).
  unsigned lds0 = (unsigned)(uintptr_t)&Vt[0][lr * 68 + lc];
  unsigned lds1 = (unsigned)(uintptr_t)&Vt[1][lr * 68 + lc];
  // Global source for j-tile jt: ht[b, jt*16 + lr, hh*64 + lc .. +3]
  unsigned long long gbase =
      (unsigned long long)(uintptr_t)(ht + (size_t)(b * kN + lr) * kHD + hh * kD + lc);
  const unsigned long long gstep = 16ull * kHD * sizeof(float);  // 16 rows

  async_load_b128(lds0, gbase);                 // prefetch tile 0
  for (int jt = 0; jt < kN / 16; ++jt) {
    int j0 = jt * 16;
    __syncthreads();  // all waves done reading buf[(jt+1)&1] (iteration jt-1)
    if (jt + 1 < kN / 16) {
      async_load_b128(((jt + 1) & 1) ? lds1 : lds0,
                      gbase + (unsigned long long)(jt + 1) * gstep);
      wait_async_le1();                         // tile jt retired (in-order)
    } else {
      wait_async_0();
    }
    __syncthreads();                            // tile jt visible to all waves
    const float* Vb = Vt[jt & 1];

    unsigned ah = arow[jt];
    float p[8];
#pragma unroll
    for (int kk = 0; kk < 4; ++kk)
#pragma unroll
      for (int u = 0; u < 2; ++u) {
        int cidx = kk * 4 + kb + u;             // local column of P tile
        float e = s_i + tg[j0 + cidx];
        e = e > 0.f ? e : 0.2f * e;
        float em = ((ah >> cidx) & 1) ? e : -1.0e9f;
        p[kk * 2 + u] = __expf(em - m_i);       // ==0 when masked & neighbors exist
      }
#pragma unroll
    for (int kk = 0; kk < 4; ++kk) {
      v2f af; af.x = p[kk * 2]; af.y = p[kk * 2 + 1];
#pragma unroll
      for (int t = 0; t < 4; ++t) {
        v2f bf;
        bf.x = Vb[(kk * 4 + kb)     * 68 + t * 16 + nn];
        bf.y = Vb[(kk * 4 + kb + 1) * 68 + t * 16 + nn];
        acc[t] = wmma_f32(af, bf, acc[t]);
      }
    }
  }
  int rb = lane >> 4;
#pragma unroll
  for (int v = 0; v < 8; ++v) {
    int row = i0 + rb * 8 + v;
    float linv = 1.0f / lrow[bh * kN + row];
#pragma unroll
    for (int t = 0; t < 4; ++t)
      out[(size_t)(bh * kN + row) * kD + t * 16 + nn] = acc[t][v] * linv;
  }
}

extern "C" void kernel_launch(void* const* d_in, const int* in_sizes, int n_in,
                              void* d_out, int out_size, void* d_ws, size_t ws_size,
                              hipStream_t stream) {
  const float* h = (const float*)d_in[0];   // [B,N,F]
  const float* W = (const float*)d_in[1];   // [F, H*D]
  const float* a = (const float*)d_in[2];   // [H, 2D, 1]
  float* out = (float*)d_out;               // [B,H,N,D] flat == reference reshape

  char* ws = (char*)d_ws;
  size_t off = 0;
  float* ht   = (float*)(ws + off); off += (size_t)kB * kN * kHD * 4;   // 16 MB
  float* inv  = (float*)(ws + off); off += (size_t)kB * kN * 4;
  float* src  = (float*)(ws + off); off += (size_t)kBH * kN * 4;
  float* tgt  = (float*)(ws + off); off += (size_t)kBH * kN * 4;
  unsigned short* adjh = (unsigned short*)(ws + off);
  off += (size_t)kB * kN * kNW * 2;                                     // 1 MB
  float* mrow = (float*)(ws + off); off += (size_t)kBH * kN * 4;
  float* lrow = (float*)(ws + off); off += (size_t)kBH * kN * 4;
  (void)ws_size; (void)in_sizes; (void)n_in; (void)out_size;

  k_norm  <<<kB * kN / 8, 256, 0, stream>>>(h, inv);
  k_proj  <<<dim3((kB * kN) / 64, kHD / 128), 256, 0, stream>>>(h, W, ht);
  k_srctgt<<<kBH * kN / 8, 256, 0, stream>>>(ht, a, src, tgt);
  k_adj   <<<kB * (kN / 16) * (kN / 16) / 8, 256, 0, stream>>>(h, inv, adjh);
  k_stats <<<kBH * kN / 8, 256, 0, stream>>>(src, tgt, adjh, mrow, lrow);
  k_attn  <<<kBH * 8, 256, 0, stream>>>(ht, src, tgt, adjh, mrow, lrow, out);
}